// BroadcastTC_35777077575987
// MI455X (gfx1250) — compile-verified
//
#include <hip/hip_runtime.h>

typedef float v2f __attribute__((ext_vector_type(2)));
typedef float v8f __attribute__((ext_vector_type(8)));

// ---------- compile-time gather tables ----------
// Slot s = c*2 + j holds (K = 4c + 2*half + j) for chunk c, vgpr j.
// iA/iB: BYTE offset (idx*4) into the 27-float tile; mA/mB: validity lane masks.
struct GatherTab {
  unsigned char iA[6][32];
  unsigned char iB[6][32];
  unsigned int  mA[6];
  unsigned int  mB[6];
};

// A rows: 0-8 out1 (T1[m*3+K], K<3) | 9-11 out2 (T1[(m-9)*9+K], K<9)
//         12,13 out3 (T1[K], T1[12+K]) | 14 out3 (T1[24+K], K<3) | 15 zero
constexpr void idxA_ct(int m, int K, int& idx, bool& v) {
  if (m < 9)        { v = K < 3;  idx = v ? m * 3 + K : 0; }
  else if (m < 12)  { v = K < 9;  idx = v ? (m - 9) * 9 + K : 0; }
  else if (m == 12) { v = true;   idx = K; }
  else if (m == 13) { v = true;   idx = 12 + K; }
  else if (m == 14) { v = K < 3;  idx = v ? 24 + K : 0; }
  else              { v = false;  idx = 0; }
}
// B cols mirrored on the T2 tile.
constexpr void idxB_ct(int n, int K, int& idx, bool& v) {
  if (n < 9)        { v = K < 3;  idx = v ? K * 9 + n : 0; }
  else if (n < 12)  { v = K < 9;  idx = v ? K * 3 + (n - 9) : 0; }
  else if (n == 12) { v = true;   idx = K; }
  else if (n == 13) { v = true;   idx = 12 + K; }
  else if (n == 14) { v = K < 3;  idx = v ? 24 + K : 0; }
  else              { v = false;  idx = 0; }
}

constexpr GatherTab makeTab() {
  GatherTab t{};
  for (int lane = 0; lane < 32; ++lane) {
    const int half = lane >> 4;
    const int r    = lane & 15;
    for (int c = 0; c < 3; ++c)
      for (int j = 0; j < 2; ++j) {
        const int s = c * 2 + j;
        const int K = 4 * c + 2 * half + j;
        int idx = 0; bool v = false;
        idxA_ct(r, K, idx, v);
        t.iA[s][lane] = (unsigned char)(idx * 4);
        if (v) t.mA[s] |= (1u << lane);
        idxB_ct(r, K, idx, v);
        t.iB[s][lane] = (unsigned char)(idx * 4);
        if (v) t.mB[s] |= (1u << lane);
      }
  }
  return t;
}

__device__ __constant__ GatherTab g_tab = makeTab();

__device__ __forceinline__ float rdlane(float v, int lane) {
  return __builtin_bit_cast(float,
      __builtin_amdgcn_readlane(__builtin_bit_cast(int, v), lane));
}

__device__ __forceinline__ float ldoff(const float* __restrict__ base, int byteOff) {
  return *(const float*)((const char*)base + byteOff);
}

__global__ void btc_wmma_kernel(const float* __restrict__ T1,
                                const float* __restrict__ T2,
                                float* __restrict__ out,
                                int npairs) {
  const int lane = (int)(threadIdx.x & 31u);
  const int half = lane >> 4;
  const int r    = lane & 15;

  const int wavesPerBlock = (int)(blockDim.x >> 5);
  const int wib = __builtin_amdgcn_readfirstlane((int)(threadIdx.x >> 5));
  const int wid = (int)blockIdx.x * wavesPerBlock + wib;
  const int nw  = (int)gridDim.x * wavesPerBlock;

  // ---- branch-free prologue: table loads + mask bit-tests ----
  int   oA[6], oB[6];
  float wA[6], wB[6];
#pragma unroll
  for (int s = 0; s < 6; ++s) {
    oA[s] = (int)g_tab.iA[s][lane];
    oB[s] = (int)g_tab.iB[s][lane];
    wA[s] = ((g_tab.mA[s] >> lane) & 1u) ? 1.0f : 0.0f;
    wB[s] = ((g_tab.mB[s] >> lane) & 1u) ? 1.0f : 0.0f;
  }

  const bool doOut1Lo = (half == 0) && (r < 9);
  const bool doOut1Hi = (half == 1) && (r < 9);
  const bool doOut2   = (half == 1) && (r >= 9) && (r < 12);
  const int  u        = r - 9;

  float* __restrict__ out1 = out;                        // npairs*81
  float* __restrict__ out2 = out + (size_t)npairs * 81;  // npairs*9
  float* __restrict__ out3 = out + (size_t)npairs * 90;  // npairs

  const float s1 = 0.57735026918962576f;  // 1/sqrt(3)
  const float s2 = 0.33333333333333333f;  // 1/3
  const float s3 = 0.19245008972987526f;  // 1/sqrt(27)

  for (int p = wid; p < npairs; p += nw) {
    const float* __restrict__ t1 = T1 + (size_t)p * 27;  // uniform base
    const float* __restrict__ t2 = T2 + (size_t)p * 27;  // uniform base

    if (p + nw < npairs) {
      __builtin_prefetch(T1 + (size_t)(p + nw) * 27, 0, 3);
      __builtin_prefetch(T2 + (size_t)(p + nw) * 27, 0, 3);
    }

    v8f acc = {0.f, 0.f, 0.f, 0.f, 0.f, 0.f, 0.f, 0.f};

#pragma unroll
    for (int c = 0; c < 3; ++c) {
      v2f a, b;
      a.x = ldoff(t1, oA[c * 2 + 0]) * wA[c * 2 + 0];
      a.y = ldoff(t1, oA[c * 2 + 1]) * wA[c * 2 + 1];
      b.x = ldoff(t2, oB[c * 2 + 0]) * wB[c * 2 + 0];
      b.y = ldoff(t2, oB[c * 2 + 1]) * wB[c * 2 + 1];
      acc = __builtin_amdgcn_wmma_f32_16x16x4_f32(
          false, a, false, b, (short)0, acc, false, false);
    }

    // ---- out1: D[0..8][0..8] * 1/sqrt(3) ----
    float* __restrict__ o1 = out1 + (size_t)p * 81;
    if (doOut1Lo) {
#pragma unroll
      for (int m = 0; m < 8; ++m) o1[m * 9 + r] = acc[m] * s1;
    }
    if (doOut1Hi) {
      o1[72 + r] = acc[0] * s1;  // row 8 = VGPR0 of lanes 16..31
    }

    // ---- out2: D[9+x][9+u] * 1/3 ----
    if (doOut2) {
      float* __restrict__ o2 = out2 + (size_t)p * 9;
      o2[u]     = acc[1] * s2;
      o2[3 + u] = acc[2] * s2;
      o2[6 + u] = acc[3] * s2;
    }

    // ---- out3: (D[12][12] + D[13][13] + D[14][14]) * 1/sqrt(27) ----
    const float d0 = rdlane(acc[4], 28);
    const float d1 = rdlane(acc[5], 29);
    const float d2 = rdlane(acc[6], 30);
    if (lane == 0) out3[p] = (d0 + d1 + d2) * s3;
  }
}

extern "C" void kernel_launch(void* const* d_in, const int* in_sizes, int n_in,
                              void* d_out, int out_size, void* d_ws, size_t ws_size,
                              hipStream_t stream) {
  const float* T1 = (const float*)d_in[0];
  const float* T2 = (const float*)d_in[1];
  float* out = (float*)d_out;

  const int npairs = in_sizes[0] / 27;  // B*C

  const int block = 256;                // 8 wave32 per workgroup
  const int wavesPerBlock = block / 32;
  int targetWaves = 32768;              // ~32 pairs per wave
  int grid = (targetWaves + wavesPerBlock - 1) / wavesPerBlock;
  int maxGrid = (npairs + wavesPerBlock - 1) / wavesPerBlock;
  if (grid > maxGrid) grid = maxGrid;
  if (grid < 1) grid = 1;

  btc_wmma_kernel<<<grid, block, 0, stream>>>(T1, T2, out, npairs);
}